// yolo_detect_target_27556510171640
// MI455X (gfx1250) — compile-verified
//
#include <hip/hip_runtime.h>

// Problem constants (from the reference)
#define NROWS   262144
#define NCLS    80
#define KROWS   (NROWS / 2)   // K = int(N * 0.5) = 131072
#define CONF_T  0.25f

typedef float v2f __attribute__((ext_vector_type(2)));
typedef float v8f __attribute__((ext_vector_type(8)));

// Launch geometry (compile-time, chosen so pass 2 divides exactly).
#define B1 256
#define T1 256
#define B2 128
#define T2 256
#define TOTAL_WAVES2 ((B2 * T2) / 32)            // 1024
#define NCHUNKS      (KROWS / 64)                // 2048 chunks of 64 scores
#define ITERS2       (NCHUNKS / TOTAL_WAVES2)    // exactly 2
static_assert(KROWS % 64 == 0, "chunking");
static_assert(NCHUNKS % TOTAL_WAVES2 == 0, "exact division -> no range guard");

// --------------------------------------------------------------------------
// Kernel 1: per-row max over 80 classes.
// 4 lanes cooperate on one 320-byte row (each lane: 5 float4 loads), combined
// with wave32 shfl_xor. 8 rows per wave. Writes scores[K] to workspace and
// atomicMin's the first row index whose max < CONF (integer atomic => exact).
// --------------------------------------------------------------------------
__global__ void rowmax_scores_kernel(const float* __restrict__ post,
                                     float* __restrict__ scores,
                                     int* __restrict__ firstBelow,
                                     int rowsPerPass)
{
    const int lane  = threadIdx.x & 31;
    const int j     = lane & 3;    // chunk id within 4-lane row team
    const int group = lane >> 2;   // row team within wave (0..7)
    const int gw    = (blockIdx.x * blockDim.x + threadIdx.x) >> 5;

    for (int r = gw * 8 + group; r < KROWS; r += rowsPerPass) {
        const float4* p = (const float4*)(post + (size_t)r * NCLS);
        // Prefetch the row this team will touch next pass (stays inside the
        // N=262144-row allocation since r + rowsPerPass < 2K <= N).
        __builtin_prefetch((const void*)(post + (size_t)(r + rowsPerPass) * NCLS), 0, 1);

        float m = -3.402823466e38f;
        #pragma unroll
        for (int c = 0; c < 5; ++c) {            // lane j reads chunks j, j+4, ...
            float4 v = p[j + 4 * c];             // global_load_b128
            m = fmaxf(m, fmaxf(fmaxf(v.x, v.y), fmaxf(v.z, v.w)));
        }
        // Combine the 4 team lanes (teams of 4 are always active together).
        m = fmaxf(m, __shfl_xor(m, 1, 32));
        m = fmaxf(m, __shfl_xor(m, 2, 32));

        if (j == 0) {
            scores[r] = m;
            if (m < CONF_T) atomicMin(firstBelow, r);
        }
    }
}

// --------------------------------------------------------------------------
// Kernel 2: masked prefix sum of scores via V_WMMA_F32_16X16X4_F32.
// A (16x4 f32, 2 VGPRs/lane) carries 64 scores per wave-iteration; B = ones,
// so D[m][n] = sum_k A[m][k] + C[m][n]: every column of C accumulates the
// per-row sums. Indices are in range by construction (static_assert above),
// so loads are unconditional global_load_b64; the prefix mask (i < firstBelow)
// is a branchless v_cndmask pair, keeping EXEC all-1s as WMMA requires.
// Deterministic per-block partials, no float atomics.
// --------------------------------------------------------------------------
__global__ void wmma_prefix_sum_kernel(const float* __restrict__ scores,
                                       const int* __restrict__ firstBelow,
                                       float* __restrict__ partials)
{
    __shared__ float lds[8];
    const int lane = threadIdx.x & 31;
    const int wib  = threadIdx.x >> 5;
    const int gw   = (blockIdx.x * blockDim.x + threadIdx.x) >> 5;
    const int fb   = *firstBelow;   // uniform; sentinel 0x7F7F7F7F if none below

    const float2* __restrict__ sp = (const float2*)scores;

    v8f c = {0.f, 0.f, 0.f, 0.f, 0.f, 0.f, 0.f, 0.f};
    const v2f bones = {1.f, 1.f};

    #pragma unroll
    for (int it = 0; it < ITERS2; ++it) {
        const int chunk = gw + it * TOTAL_WAVES2;         // < NCHUNKS always
        const int pair  = chunk * 32 + lane;              // float2 index
        const int i0    = pair * 2;                       // < KROWS always
        float2 v = sp[pair];                              // global_load_b64
        float s0 = (i0     < fb) ? v.x : 0.f;             // prefix mask
        float s1 = (i0 + 1 < fb) ? v.y : 0.f;
        v2f a = {s0, s1};
        // 8 args: (neg_a, A, neg_b, B, c_mod, C, reuse_a, reuse_b)
        c = __builtin_amdgcn_wmma_f32_16x16x4_f32(false, a, false, bones,
                                                  (short)0, c, false, false);
    }

    // Columns of C are identical; lane L holds M=0..7 (VGPRs 0..7) of column
    // L%16, lane L^16 holds M=8..15. Wave total = own half + partner half.
    float s = c[0] + c[1] + c[2] + c[3] + c[4] + c[5] + c[6] + c[7];
    s += __shfl_xor(s, 16, 32);

    if (lane == 0) lds[wib] = s;
    __syncthreads();
    if (threadIdx.x == 0) {
        float t = 0.f;
        #pragma unroll
        for (int w = 0; w < 8; ++w) t += lds[w];   // fixed order => deterministic
        partials[blockIdx.x] = t;
    }
}

// --------------------------------------------------------------------------
// Kernel 3: deterministic serial sum of the per-block partials.
// --------------------------------------------------------------------------
__global__ void final_sum_kernel(const float* __restrict__ partials,
                                 int n, float* __restrict__ out)
{
    if (threadIdx.x == 0 && blockIdx.x == 0) {
        float t = 0.f;
        for (int i = 0; i < n; ++i) t += partials[i];
        out[0] = t;
    }
}

extern "C" void kernel_launch(void* const* d_in, const int* in_sizes, int n_in,
                              void* d_out, int out_size, void* d_ws, size_t ws_size,
                              hipStream_t stream) {
    const float* post = (const float*)d_in[0];   // [N, 80] f32
    // d_in[1] (pre_post_boxes) is unused by the 'class' output path.
    float* out = (float*)d_out;                  // single f32 scalar

    char*  ws         = (char*)d_ws;
    int*   firstBelow = (int*)ws;                         // 4 bytes
    float* scores     = (float*)(ws + 256);               // K floats (512 KB)
    float* partials   = (float*)(ws + 256 + (size_t)KROWS * 4);

    // Sentinel 0x7F7F7F7F (> K) for "no row below threshold". Graph-capturable.
    hipMemsetAsync(firstBelow, 0x7F, 4, stream);

    // Pass 1: row maxes. 256 blocks x 256 threads = 2048 waves, 8 rows/wave
    // per pass -> 16384 rows/pass -> exactly 8 passes over K=131072.
    const int rowsPerPass = ((B1 * T1) >> 5) * 8;
    rowmax_scores_kernel<<<B1, T1, 0, stream>>>(post, scores, firstBelow, rowsPerPass);

    // Pass 2: WMMA masked sum. 128 blocks x 256 threads = 1024 waves,
    // 64 scores per wave-iteration -> exactly 2 iterations each, no guards.
    wmma_prefix_sum_kernel<<<B2, T2, 0, stream>>>(scores, firstBelow, partials);

    // Pass 3: fold 128 partials deterministically.
    final_sum_kernel<<<1, 32, 0, stream>>>(partials, B2, out);
}